// AdditiveAttention_24000277250426
// MI455X (gfx1250) — compile-verified
//
#include <hip/hip_runtime.h>
#include <hip/hip_bf16.h>

// ---- problem constants (from reference) ----
#define Bc   2
#define Tc   512      // T == S
#define Dc   512      // input feature dim
#define UN   512      // UNITS
#define Hc   8
#define DEP  64       // UNITS / H
#define ROWS (Bc*Tc)  // 1024
#define TQ   8        // queries per block in the fused attention kernel

typedef __attribute__((ext_vector_type(2))) float v2f;
typedef __attribute__((ext_vector_type(8))) float v8f;

// Native gfx1250 V_TANH_F32 (TRANS pipe, co-executes with VALU/WMMA).
__device__ __forceinline__ float tanh_hw(float x) {
#if __has_builtin(__builtin_amdgcn_tanhf)
    return __builtin_amdgcn_tanhf(x);
#elif __has_builtin(__builtin_amdgcn_tanh_f32)
    return __builtin_amdgcn_tanh_f32(x);
#else
    float r;
    // ISA 7.4.3: 1 independent op or V_NOP required after a TRANS op
    // before its output is consumed; the asm block hides the dep from
    // the scheduler, so insert the V_NOP ourselves.
    asm("v_tanh_f32 %0, %1\n\tv_nop" : "=v"(r) : "v"(x));
    return r;
#endif
}

// ---------------------------------------------------------------------------
// Kernel 1: Q = query@Wq+bq, K = key@Wk+bk, written head-split [B,H,T,64].
// One wave computes one 16x16 output tile with V_WMMA_F32_16X16X4_F32.
// ---------------------------------------------------------------------------
__global__ void k_proj_qk(const float* __restrict__ qin, const float* __restrict__ kin,
                          const float* __restrict__ Wq, const float* __restrict__ bq,
                          const float* __restrict__ Wk, const float* __restrict__ bk,
                          float* __restrict__ Qh, float* __restrict__ Kh) {
    const int lane = threadIdx.x & 31;
    const int wid  = blockIdx.x * 8 + (threadIdx.x >> 5);
    const int mat  = wid >> 11;            // 0 = Q path, 1 = K path
    const int tile = wid & 2047;
    const int tm   = tile >> 5;            // 0..63  (row tile over 1024 rows)
    const int tn   = tile & 31;            // 0..31  (col tile over 512 cols)

    const float* A    = (mat ? kin : qin) + (tm * 16) * Dc;
    const float* W    = (mat ? Wk  : Wq ) + tn * 16;
    const float* bias = (mat ? bk  : bq );
    float*       Out  = (mat ? Kh  : Qh );

    const int half = lane >> 4;            // 0: K={0,1}, 1: K={2,3} sub-columns
    const int m    = lane & 15;

    v8f c = {};
    #pragma unroll 4
    for (int kk = 0; kk < Dc; kk += 4) {
        const int ka = kk + 2 * half;
        v2f a, bfr;
        a.x   = A[m * Dc + ka];
        a.y   = A[m * Dc + ka + 1];
        bfr.x = W[ka * UN + m];
        bfr.y = W[(ka + 1) * UN + m];
        c = __builtin_amdgcn_wmma_f32_16x16x4_f32(false, a, false, bfr,
                                                  (short)0, c, false, false);
    }

    const int n  = tn * 16 + m;
    const float bv = bias[n];
    const int h  = n >> 6, d = n & 63;
    #pragma unroll
    for (int i = 0; i < 8; ++i) {
        const int r = tm * 16 + i + 8 * half;   // C layout: VGPR i -> M=i (lo) / i+8 (hi)
        const int bb = r >> 9, t = r & 511;
        Out[((bb * Hc + h) * Tc + t) * DEP + d] = c[i] + bv;
    }
}

// ---------------------------------------------------------------------------
// Kernel 2: per-head maps  q' = Q[b,h] @ Wq_h[h],  k' = K[b,h] @ Wk_h[h] + b_h[h]
// ---------------------------------------------------------------------------
__global__ void k_head_proj(const float* __restrict__ Qh, const float* __restrict__ Kh,
                            const float* __restrict__ Wqh, const float* __restrict__ Wkh,
                            const float* __restrict__ bh,
                            float* __restrict__ qo, float* __restrict__ ko) {
    const int lane = threadIdx.x & 31;
    const int wid  = blockIdx.x * 8 + (threadIdx.x >> 5);
    const int mat  = wid >> 11;            // 0 = q', 1 = k'
    const int tile = wid & 2047;
    const int tm   = tile >> 2;            // 0..511 row tiles (8192 rows)
    const int tn   = tile & 3;             // 0..3   col tiles (64 cols)
    const int row0 = tm * 16;
    const int h    = (row0 >> 9) & (Hc - 1);

    const float* A = (mat ? Kh  : Qh ) + row0 * DEP;
    const float* W = (mat ? Wkh : Wqh) + h * DEP * DEP + tn * 16;
    float*     Out = (mat ? ko  : qo );

    const int half = lane >> 4;
    const int m    = lane & 15;

    v8f c = {};
    #pragma unroll
    for (int kk = 0; kk < DEP; kk += 4) {
        const int ka = kk + 2 * half;
        v2f a, bfr;
        a.x   = A[m * DEP + ka];
        a.y   = A[m * DEP + ka + 1];
        bfr.x = W[ka * DEP + m];
        bfr.y = W[(ka + 1) * DEP + m];
        c = __builtin_amdgcn_wmma_f32_16x16x4_f32(false, a, false, bfr,
                                                  (short)0, c, false, false);
    }

    const int n = tn * 16 + m;
    const float bias = mat ? bh[h * DEP + n] : 0.0f;   // fold b_h into k' only
    #pragma unroll
    for (int i = 0; i < 8; ++i) {
        const int r = row0 + i + 8 * half;
        Out[r * DEP + n] = c[i] + bias;
    }
}

// ---------------------------------------------------------------------------
// Kernel 3: fused additive score -> softmax -> attn @ K (reference uses K!).
// TQ=8 query rows per 256-thread block: each k' row is loaded once per block
// and reused across 8 queries (8x less L2 traffic than one-t-per-block).
// Per-wave softmax via __shfl_xor (wave32), native V_TANH_F32 for the score.
// ---------------------------------------------------------------------------
__global__ void k_attn(const float* __restrict__ qo, const float* __restrict__ ko,
                       const float* __restrict__ Kh, const float* __restrict__ va,
                       float* __restrict__ merged) {
    __shared__ float qv[TQ][DEP];      // 2 KB
    __shared__ float sc[TQ][Tc];       // 16 KB
    __shared__ float invs[TQ];
    const int bh  = blockIdx.x >> 6;          // b*H + h   (grid = 16 * 64 blocks)
    const int t0  = (blockIdx.x & 63) * TQ;   // first query row of this block
    const int h   = bh & (Hc - 1);
    const int tid = threadIdx.x;

    // stage q' rows for the TQ queries
    for (int i = tid; i < TQ * DEP; i += 256)
        (&qv[0][0])[i] = qo[(bh * Tc + t0) * DEP + i];
    __syncthreads();

    // v_a into registers (uniform per block, reused 1024x per thread)
    float vr[DEP];
    #pragma unroll
    for (int e = 0; e < DEP; ++e) vr[e] = va[h * DEP + e];

    // scores: thread owns keys s = tid, tid+256; k' row kept in registers
    for (int j = 0; j < 2; ++j) {
        const int s = tid + j * 256;
        const float* kr = ko + (bh * Tc + s) * DEP;
        float krr[DEP];
        #pragma unroll
        for (int e = 0; e < DEP; ++e) krr[e] = kr[e];
        for (int t = 0; t < TQ; ++t) {
            float acc = 0.f;
            #pragma unroll
            for (int e = 0; e < DEP; ++e)
                acc += vr[e] * tanh_hw(qv[t][e] + krr[e]);  // b_h folded into k'
            sc[t][s] = acc;
        }
    }
    __syncthreads();

    // per-wave softmax: wave w owns query row t=w (8 waves == TQ)
    {
        const int w = tid >> 5, l = tid & 31;
        float mx = -3.402823466e38f;
        #pragma unroll
        for (int i = 0; i < Tc / 32; ++i) mx = fmaxf(mx, sc[w][l + i * 32]);
        #pragma unroll
        for (int off = 16; off > 0; off >>= 1)
            mx = fmaxf(mx, __shfl_xor(mx, off, 32));
        float sum = 0.f;
        #pragma unroll
        for (int i = 0; i < Tc / 32; ++i) {
            const float e = __expf(sc[w][l + i * 32] - mx);
            sc[w][l + i * 32] = e;
            sum += e;
        }
        #pragma unroll
        for (int off = 16; off > 0; off >>= 1)
            sum += __shfl_xor(sum, off, 32);
        if (l == 0) invs[w] = 1.f / sum;
    }
    __syncthreads();

    // attn @ K: thread owns column d for t in {tg, tg+4}; K element read once,
    // used for two query rows.
    const int d = tid & 63, tg = tid >> 6;
    const float* Kb = Kh + bh * Tc * DEP + d;
    float a0 = 0.f, a1 = 0.f;
    for (int s = 0; s < Tc; ++s) {
        const float kv = Kb[s * DEP];
        a0 += sc[tg][s] * kv;
        a1 += sc[tg + 4][s] * kv;
    }
    const int b = bh >> 3;
    merged[(b * Tc + t0 + tg    ) * UN + h * DEP + d] = a0 * invs[tg];
    merged[(b * Tc + t0 + tg + 4) * UN + h * DEP + d] = a1 * invs[tg + 4];
}

// ---------------------------------------------------------------------------
// Kernel 4: output projection  out = merged(1024x512) @ Wo + bo
// ---------------------------------------------------------------------------
__global__ void k_out_proj(const float* __restrict__ M, const float* __restrict__ Wo,
                           const float* __restrict__ bo, float* __restrict__ out) {
    const int lane = threadIdx.x & 31;
    const int wid  = blockIdx.x * 8 + (threadIdx.x >> 5);
    const int tm   = wid >> 5;             // 0..63
    const int tn   = wid & 31;             // 0..31

    const float* A = M  + (tm * 16) * UN;
    const float* W = Wo + tn * 16;

    const int half = lane >> 4;
    const int m    = lane & 15;

    v8f c = {};
    #pragma unroll 4
    for (int kk = 0; kk < UN; kk += 4) {
        const int ka = kk + 2 * half;
        v2f a, bfr;
        a.x   = A[m * UN + ka];
        a.y   = A[m * UN + ka + 1];
        bfr.x = W[ka * UN + m];
        bfr.y = W[(ka + 1) * UN + m];
        c = __builtin_amdgcn_wmma_f32_16x16x4_f32(false, a, false, bfr,
                                                  (short)0, c, false, false);
    }

    const int n  = tn * 16 + m;
    const float bv = bo[n];
    #pragma unroll
    for (int i = 0; i < 8; ++i) {
        const int r = tm * 16 + i + 8 * half;
        out[r * UN + n] = c[i] + bv;
    }
}

// ---------------------------------------------------------------------------
extern "C" void kernel_launch(void* const* d_in, const int* in_sizes, int n_in,
                              void* d_out, int out_size, void* d_ws, size_t ws_size,
                              hipStream_t stream) {
    const float* query = (const float*)d_in[0];
    const float* key   = (const float*)d_in[1];
    // d_in[2] = value  (dead in the reference: V computed but unused)
    const float* Wq    = (const float*)d_in[3];
    const float* bq    = (const float*)d_in[4];
    const float* Wk    = (const float*)d_in[5];
    const float* bk    = (const float*)d_in[6];
    // d_in[7], d_in[8] = Wv, bv (unused)
    const float* Wqh   = (const float*)d_in[9];
    const float* Wkh   = (const float*)d_in[10];
    const float* vah   = (const float*)d_in[11];
    const float* bh    = (const float*)d_in[12];   // folded into k'
    const float* Wo    = (const float*)d_in[13];
    const float* bo    = (const float*)d_in[14];
    float* out = (float*)d_out;

    // workspace: 5 buffers of B*H*T*DEP = 524288 floats (2 MB each)
    const size_t NB = (size_t)Bc * Hc * Tc * DEP;
    float* Qh = (float*)d_ws;
    float* Kh = Qh + NB;
    float* qo = Kh + NB;
    float* ko = qo + NB;
    float* mg = ko + NB;   // merged heads [B, T, UNITS]

    // 1) input projections (4096 waves -> 512 blocks of 8 waves)
    k_proj_qk<<<512, 256, 0, stream>>>(query, key, Wq, bq, Wk, bk, Qh, Kh);
    // 2) per-head 64x64 maps (+ b_h fold)
    k_head_proj<<<512, 256, 0, stream>>>(Qh, Kh, Wqh, Wkh, bh, qo, ko);
    // 3) fused score/softmax/attn@K, TQ=8 queries per block
    k_attn<<<(Bc * Hc * Tc) / TQ, 256, 0, stream>>>(qo, ko, Kh, vah, mg);
    // 4) output projection (2048 waves -> 256 blocks)
    k_out_proj<<<256, 256, 0, stream>>>(mg, Wo, bo, out);
}